// Attention_70583492542709
// MI455X (gfx1250) — compile-verified
//
#include <hip/hip_runtime.h>
#include <hip/hip_bf16.h>

// ---------------------------------------------------------------------------
// Additive (Bahdanau) attention for MI455X / gfx1250.
//   B=8, T=64, S=512, K=D=512
//
//   k1: dec_proj = dec @ W_d + b_energy     (WMMA f32 16x16x4, LDS-staged A)
//   k2: enc_proj = enc @ W_e                (WMMA f32 16x16x4, natural layout)
//   k3: scores/softmax: wave-per-s, lanes across K, register-resident dp/v,
//       coalesced b128 ep loads, tanh+fma, shfl reduction, block softmax
//   k4: context = probs @ enc               (WMMA f32 16x16x4, batched)
// ---------------------------------------------------------------------------

typedef __attribute__((ext_vector_type(2))) float v2f;
typedef __attribute__((ext_vector_type(4))) float v4f;
typedef __attribute__((ext_vector_type(8))) float v8f;

#define KDIM 512          // shared inner dimension of every GEMM here
#define KP   514          // padded LDS stride (avoids 16-way bank conflicts)
#define SDIM 512
#define TLEN 64
#define BATCH 8

// ---------------------------------------------------------------------------
// C = A(Mx512) * B(512xN) [+bias], batched.
// grid: (N/64, M/16, batch)   block: 128 threads = 4 waves, one 16x16 tile each
// Fragment layouts per CDNA5 ISA 7.12.2 (half-wave K split).
// ---------------------------------------------------------------------------
__global__ __launch_bounds__(128)
void wmma_gemm512_f32(const float* __restrict__ A, const float* __restrict__ B,
                      const float* __restrict__ bias, float* __restrict__ C,
                      int N,
                      long long strideA, long long strideB, long long strideC)
{
    __shared__ float sAt[16 * KP];

    const int tid  = threadIdx.x;
    const int lane = tid & 31;
    const int wv   = tid >> 5;
    const int h    = lane >> 4;      // half-wave select (K split)
    const int idx  = lane & 15;
    const int m0   = blockIdx.y * 16;
    const int n0   = (blockIdx.x * 4 + wv) * 16;
    const int b    = blockIdx.z;

    const float* Ab = A + (long long)b * strideA + (long long)m0 * KDIM;
    const float* Bb = B + (long long)b * strideB;

    // cooperative, fully coalesced 16x512 A-tile load into padded LDS
    for (int i = tid; i < 16 * KDIM; i += 128) {
        int r  = i >> 9;             // /512
        int cc = i & (KDIM - 1);
        sAt[r * KP + cc] = Ab[(long long)r * KDIM + cc];
    }
    __syncthreads();

    v8f acc = {};
#pragma unroll 4
    for (int kk = 0; kk < KDIM; kk += 4) {
        const int k0 = kk + 2 * h;
        v2f a = *(const v2f*)&sAt[idx * KP + k0];   // 8B aligned (k0 even, KP even)
        v2f bv;
        bv.x = Bb[(long long)k0 * N + n0 + idx];
        bv.y = Bb[(long long)(k0 + 1) * N + n0 + idx];
        acc = __builtin_amdgcn_wmma_f32_16x16x4_f32(false, a, false, bv,
                                                    (short)0, acc, false, false);
    }

    if (bias) {
        float bb = bias[n0 + idx];
#pragma unroll
        for (int r = 0; r < 8; ++r) acc[r] += bb;
    }

    float* Cb = C + (long long)b * strideC;
#pragma unroll
    for (int r = 0; r < 8; ++r) {
        int m = m0 + r + 8 * h;
        Cb[(long long)m * N + n0 + idx] = acc[r];
    }
}

// ---------------------------------------------------------------------------
// tanh: prefer native gfx1250 v_tanh_f32; else branch-free exp-based version
// ---------------------------------------------------------------------------
__device__ __forceinline__ float fast_tanh(float x)
{
#if __has_builtin(__builtin_amdgcn_tanhf)
    return __builtin_amdgcn_tanhf(x);
#elif __has_builtin(__builtin_amdgcn_tanh_f32)
    return __builtin_amdgcn_tanh_f32(x);
#else
    float e = __expf(2.0f * x);          // v_exp_f32
    return 1.0f - 2.0f / (e + 1.0f);     // exact at +/-inf, monotone, branch-free
#endif
}

// ---------------------------------------------------------------------------
// scores + masked softmax.  grid: (T, B), block: 512 = 16 waves.
// Wave wv owns s in [wv*32, wv*32+32). Lane covers k = j*128 + lane*4 + c.
//   score[s] = sum_k v[k] * tanh(dp[b,t,k] + ep[b,s,k])
// dp-row and v chunks are loop-invariant per lane -> registers (no LDS reads
// in the hot loop). ep rows read as coalesced 512B global_load_b128 packets.
// ---------------------------------------------------------------------------
__global__ __launch_bounds__(512)
void scores_softmax(const float* __restrict__ dp,    // (B,T,512)  bias folded in
                    const float* __restrict__ ep,    // (B,S,512)
                    const float* __restrict__ mask,  // (B,S)
                    const float* __restrict__ v,     // (512)
                    float* __restrict__ probs)       // (B,T,S)
{
    __shared__ float red[SDIM];

    const int t    = blockIdx.x;
    const int b    = blockIdx.y;
    const int tid  = threadIdx.x;
    const int lane = tid & 31;
    const int wv   = tid >> 5;

    // register-resident per-lane chunks of dp-row (+bias already folded) and v
    const float* dprow = dp + ((long long)b * TLEN + t) * KDIM;
    v4f dpv[4], vv[4];
#pragma unroll
    for (int j = 0; j < 4; ++j) {
        dpv[j] = *(const v4f*)&dprow[j * 128 + lane * 4];
        vv[j]  = *(const v4f*)&v[j * 128 + lane * 4];
    }

    const float* epb = ep + (long long)b * SDIM * KDIM;

    for (int i = 0; i < 32; ++i) {
        const int s = wv * 32 + i;
        const float* eprow = epb + (long long)s * KDIM;
        if (i < 31)  // pull next row toward the WGP (global_prefetch_b8)
            __builtin_prefetch(&eprow[KDIM + lane * 4], 0, 0);

        float acc = 0.0f;
#pragma unroll
        for (int j = 0; j < 4; ++j) {
            v4f e = *(const v4f*)&eprow[j * 128 + lane * 4];
#pragma unroll
            for (int c = 0; c < 4; ++c)
                acc = fmaf(vv[j][c], fast_tanh(dpv[j][c] + e[c]), acc);
        }
        // wave-level sum across 32 lanes
#pragma unroll
        for (int m = 16; m > 0; m >>= 1)
            acc += __shfl_xor(acc, m, 32);
        if (lane == 0) red[s] = acc;
    }
    __syncthreads();

    // ---- block softmax over s (max, sum, mask, renorm) ----
    const int s = tid;
    float sc = red[s];
    __syncthreads();

    for (int o = SDIM / 2; o > 0; o >>= 1) {
        if (s < o) red[s] = fmaxf(red[s], red[s + o]);
        __syncthreads();
    }
    float mx = red[0]; __syncthreads();

    float e = __expf(sc - mx);
    red[s] = e; __syncthreads();
    for (int o = SDIM / 2; o > 0; o >>= 1) {
        if (s < o) red[s] += red[s + o];
        __syncthreads();
    }
    float se = red[0]; __syncthreads();

    float p = (e / se) * mask[(long long)b * SDIM + s];
    red[s] = p; __syncthreads();
    for (int o = SDIM / 2; o > 0; o >>= 1) {
        if (s < o) red[s] += red[s + o];
        __syncthreads();
    }
    float sp = red[0];

    probs[((long long)b * TLEN + t) * SDIM + s] = p / (sp + 1e-12f);
}

// ---------------------------------------------------------------------------
extern "C" void kernel_launch(void* const* d_in, const int* in_sizes, int n_in,
                              void* d_out, int out_size, void* d_ws, size_t ws_size,
                              hipStream_t stream)
{
    const float* dec  = (const float*)d_in[0];   // (8,64,512)
    const float* enc  = (const float*)d_in[1];   // (8,512,512)
    const float* msk  = (const float*)d_in[2];   // (8,512)
    const float* W    = (const float*)d_in[3];   // (1024,512) row-major
    const float* be   = (const float*)d_in[4];   // (512)
    const float* v    = (const float*)d_in[5];   // (512)

    float* ctx   = (float*)d_out;                        // (8,64,512)
    float* probs = ctx + (long long)BATCH * TLEN * KDIM; // (8,64,512)

    float* dp  = (float*)d_ws;                              // 1 MB  (B,T,K)
    float* epj = dp + (long long)BATCH * TLEN * KDIM;       // 8 MB  (B,S,K)

    const float* Wd = W;                 // rows [0,512)
    const float* We = W + 512LL * 512;   // rows [512,1024)

    // 1) dec_proj (+bias): M = B*T = 512, N = 512, single "batch"
    wmma_gemm512_f32<<<dim3(8, 32, 1), 128, 0, stream>>>(
        dec, Wd, be, dp, 512, 0LL, 0LL, 0LL);

    // 2) enc_proj: per batch M = S = 512, N = 512, natural (B,S,K) layout
    wmma_gemm512_f32<<<dim3(8, 32, BATCH), 128, 0, stream>>>(
        enc, We, nullptr, epj, 512,
        512LL * 512, 0LL, 512LL * 512);

    // 3) scores + masked softmax -> probs
    scores_softmax<<<dim3(TLEN, BATCH), 512, 0, stream>>>(dp, epj, msk, v, probs);

    // 4) context = probs @ enc : per batch M = T = 64, N = D = 512, K = S = 512
    wmma_gemm512_f32<<<dim3(8, 4, BATCH), 128, 0, stream>>>(
        probs, enc, nullptr, ctx, 512,
        (long long)TLEN * SDIM, 512LL * 512, (long long)TLEN * KDIM);
}